// Flow_65171833750030
// MI455X (gfx1250) — compile-verified
//
#include <hip/hip_runtime.h>
#include <math.h>

#define NLAYERS 10
#define RPB 256                       // rows per block (8 wave32s)
#define AFF_FLOATS (NLAYERS * 2 * 3)  // 60 floats / row
#define AFF_BYTES  (AFF_FLOATS * 4)   // 240 bytes / row
#define CHUNKS     (AFF_BYTES / 16)   // 15 x b128 chunks / row

__device__ __forceinline__ float fexp2(float x) { return __builtin_amdgcn_exp2f(x); } // 2^x
__device__ __forceinline__ float flog2(float x) { return __builtin_amdgcn_logf(x); }  // log2(x)
__device__ __forceinline__ float frcp (float x) { return __builtin_amdgcn_rcpf(x); }

__global__ __launch_bounds__(RPB) void Flow_main_kernel(
    const float* __restrict__ x,
    const float* __restrict__ aff,
    const float* __restrict__ Ws,
    const float* __restrict__ bs,
    const float* __restrict__ lW,
    const float* __restrict__ lb,
    const float* __restrict__ ldc_p,
    float* __restrict__ out_z,
    float* __restrict__ out_ld,
    int N)
{
    __shared__ __align__(16) float smem[RPB * AFF_FLOATS];  // 61440 B tile
    const int tid  = threadIdx.x;
    const int row0 = blockIdx.x * RPB;
    int rows = N - row0;
    if (rows > RPB) rows = RPB;
    const float* gsrc = aff + (size_t)row0 * AFF_FLOATS;

    // ---- Stage affine tile into LDS, coalesced, via CDNA5 async global->LDS DMA ----
#if defined(__AMDGCN__) && __has_builtin(__builtin_amdgcn_global_load_async_to_lds_b128)
    {
        typedef int v4i __attribute__((ext_vector_type(4)));
        typedef __attribute__((address_space(1))) v4i glb_v4i;
        typedef __attribute__((address_space(3))) v4i lds_v4i;
        glb_v4i* gp = (glb_v4i*)gsrc;   // global src
        lds_v4i* lp = (lds_v4i*)smem;   // LDS dst
        if (rows == RPB) {
            // Full tile: issue all 15 async b128 copies back-to-back.
#pragma unroll
            for (int i = 0; i < CHUNKS; ++i) {
                const int j = tid + i * RPB;
                __builtin_amdgcn_global_load_async_to_lds_b128(gp + j, lp + j, 0, 0);
            }
        } else {
            const int nchunk = rows * CHUNKS;
            for (int j = tid; j < nchunk; j += RPB)
                __builtin_amdgcn_global_load_async_to_lds_b128(gp + j, lp + j, 0, 0);
        }
#if __has_builtin(__builtin_amdgcn_s_wait_asynccnt)
        __builtin_amdgcn_s_wait_asynccnt(0);
#else
        asm volatile("s_wait_asynccnt 0" ::: "memory");
#endif
    }
#else
    {
        const int nchunk = rows * CHUNKS;
        for (int j = tid; j < nchunk; j += RPB)
            ((float4*)smem)[j] = ((const float4*)gsrc)[j];
    }
#endif
    __syncthreads();

    if (tid < rows) {
        const int row = row0 + tid;
        const float* a = smem + tid * AFF_FLOATS;
        const float ldc = ldc_p[0];               // uniform slogdet constant

        float z0 = x[(size_t)row * 3 + 0];
        float z1 = x[(size_t)row * 3 + 1];
        float z2 = x[(size_t)row * 3 + 2];
        float ld = 0.0f;

        const float L2E  = 1.4426950408889634f;   // log2(e)
        const float N2L  = -2.8853900817779268f;  // -2*log2(e)
        const float LN2  = 0.6931471805599453f;

#pragma unroll
        for (int i = 0; i < NLAYERS; ++i) {
            const float* W = Ws + i * 9;          // uniform -> scalar loads
            const float* b = bs + i * 3;
            float t0 = fmaf(z2, W[6], fmaf(z1, W[3], fmaf(z0, W[0], b[0])));
            float t1 = fmaf(z2, W[7], fmaf(z1, W[4], fmaf(z0, W[1], b[1])));
            float t2 = fmaf(z2, W[8], fmaf(z1, W[5], fmaf(z0, W[2], b[2])));

            float s0 = a[i * 6 + 0], s1 = a[i * 6 + 1], s2 = a[i * 6 + 2];
            float h0 = a[i * 6 + 3], h1 = a[i * 6 + 4], h2 = a[i * 6 + 5];

            // sigmoid(s+2)+eps : exp(-(s+2)) = 2^(s*-log2e - 2*log2e)
            float sc0 = frcp(1.0f + fexp2(fmaf(s0, -L2E, N2L))) + 1e-4f;
            float sc1 = frcp(1.0f + fexp2(fmaf(s1, -L2E, N2L))) + 1e-4f;
            float sc2 = frcp(1.0f + fexp2(fmaf(s2, -L2E, N2L))) + 1e-4f;

            z0 = fmaf(t0, sc0, h0);
            z1 = fmaf(t1, sc1, h1);
            z2 = fmaf(t2, sc2, h2);
            // sum(log(scale)) = ln2 * log2(sc0*sc1*sc2)
            ld = fmaf(flog2(sc0 * sc1 * sc2), LN2, ld);
        }

        float t0 = fmaf(z2, lW[6], fmaf(z1, lW[3], fmaf(z0, lW[0], lb[0])));
        float t1 = fmaf(z2, lW[7], fmaf(z1, lW[4], fmaf(z0, lW[1], lb[1])));
        float t2 = fmaf(z2, lW[8], fmaf(z1, lW[5], fmaf(z0, lW[2], lb[2])));

        ld += ldc;                                        // slogdet constants
        ld -= 0.5f * (t0 * t0 + t1 * t1 + t2 * t2);
        ld -= 2.7568155996140185f;                        // 1.5 * log(2*pi)

        out_z[(size_t)row * 3 + 0] = t0;
        out_z[(size_t)row * 3 + 1] = t1;
        out_z[(size_t)row * 3 + 2] = t2;
        out_ld[row] = ld;
    }
}

// Row-independent constant: sum of log|det W_i| + log|det last_W|.
// (Fully scalarized by the compiler: s_mul_f32 / s_fmac_f32 / v_s_log_f32.)
__global__ void Flow_ldconst_kernel(const float* __restrict__ Ws,
                                    const float* __restrict__ lW,
                                    float* __restrict__ outc) {
    float acc = 0.0f;
    for (int i = 0; i <= NLAYERS; ++i) {
        const float* W = (i < NLAYERS) ? (Ws + i * 9) : lW;
        float det = W[0] * (W[4] * W[8] - W[5] * W[7])
                  - W[1] * (W[3] * W[8] - W[5] * W[6])
                  + W[2] * (W[3] * W[7] - W[4] * W[6]);
        acc += logf(fabsf(det));
    }
    outc[0] = acc;
}

extern "C" void kernel_launch(void* const* d_in, const int* in_sizes, int n_in,
                              void* d_out, int out_size, void* d_ws, size_t ws_size,
                              hipStream_t stream) {
    (void)n_in; (void)out_size; (void)ws_size;
    const float* x   = (const float*)d_in[0];
    const float* aff = (const float*)d_in[1];
    const float* Ws  = (const float*)d_in[2];
    const float* bs  = (const float*)d_in[3];
    const float* lW  = (const float*)d_in[4];
    const float* lb  = (const float*)d_in[5];
    const int N = in_sizes[0] / 3;

    float* ldc    = (float*)d_ws;
    float* out_z  = (float*)d_out;
    float* out_ld = out_z + (size_t)N * 3;

    Flow_ldconst_kernel<<<1, 1, 0, stream>>>(Ws, lW, ldc);
    int blocks = (N + RPB - 1) / RPB;
    Flow_main_kernel<<<blocks, RPB, 0, stream>>>(x, aff, Ws, bs, lW, lb, ldc,
                                                 out_z, out_ld, N);
}